// Att_cat_norm_inte_42984032698903
// MI455X (gfx1250) — compile-verified
//
#include <hip/hip_runtime.h>

typedef __attribute__((ext_vector_type(2))) float v2f;
typedef __attribute__((ext_vector_type(8))) float v8f;

#define B_ 32
#define N_ 128
#define D_ 64
#define LN_EPS_ 1e-5f
#define SLOPE_ 0.01f

// ---------------------------------------------------------------------------
// Kernel 1: per-row LayerNorm of embeddings + q/k projections.
// One wave (32 lanes) per row of 64 floats (2 per lane). Pure shuffle reduce.
// ---------------------------------------------------------------------------
__global__ __launch_bounds__(128) void rowstats_kernel(
    const float* __restrict__ e, const float* __restrict__ gamma,
    const float* __restrict__ beta, const float* __restrict__ att_w,
    float* __restrict__ qk) {
  int wave = threadIdx.x >> 5, lane = threadIdx.x & 31;
  int row = blockIdx.x * 4 + wave;  // < B_*N_
  float2 v = ((const float2*)(e + (size_t)row * D_))[lane];
  float s = v.x + v.y;
  #pragma unroll
  for (int o = 16; o; o >>= 1) s += __shfl_xor(s, o, 32);
  float mu = s * (1.0f / D_);
  float dx = v.x - mu, dy = v.y - mu;
  float ss = dx * dx + dy * dy;
  #pragma unroll
  for (int o = 16; o; o >>= 1) ss += __shfl_xor(ss, o, 32);
  float rstd = rsqrtf(ss * (1.0f / D_) + LN_EPS_);
  float2 g  = ((const float2*)gamma)[lane];
  float2 bt = ((const float2*)beta)[lane];
  float l0 = dx * rstd * g.x + bt.x;
  float l1 = dy * rstd * g.y + bt.y;
  float2 wq = ((const float2*)att_w)[lane];
  float2 wk = ((const float2*)(att_w + D_))[lane];
  float q = l0 * wq.x + l1 * wq.y;
  float k = l0 * wk.x + l1 * wk.y;
  #pragma unroll
  for (int o = 16; o; o >>= 1) {
    q += __shfl_xor(q, o, 32);
    k += __shfl_xor(k, o, 32);
  }
  if (lane == 0) {
    qk[row] = q;
    qk[B_ * N_ + row] = k;
  }
}

// ---------------------------------------------------------------------------
// Kernel 2: stream value[b,i,j,d] = e[b,i,d]*e[b,j,d].  134 MB of B128 stores;
// this is the bandwidth-critical kernel (reads hit L2: embeddings are 1 MB).
// ---------------------------------------------------------------------------
__global__ __launch_bounds__(256) void value_outer_kernel(
    const float* __restrict__ e, float* __restrict__ value) {
  unsigned idx = blockIdx.x * 256u + threadIdx.x;  // one float4 each
  unsigned dq  = idx & 15u;                        // D/4 = 16 float4 per row
  unsigned bij = idx >> 4;
  unsigned j   = bij & (N_ - 1);
  unsigned bi  = bij >> 7;
  unsigned i   = bi & (N_ - 1);
  unsigned b   = bi >> 7;
  const float4* e4 = (const float4*)e;
  float4 ei = e4[(b * N_ + i) * (D_ / 4) + dq];
  float4 ej = e4[(b * N_ + j) * (D_ / 4) + dq];
  float4 v;
  v.x = ei.x * ej.x; v.y = ei.y * ej.y;
  v.z = ei.z * ej.z; v.w = ei.w * ej.w;
  ((float4*)value)[idx] = v;
}

// ---------------------------------------------------------------------------
// Kernel 3: three fp32 WMMA Gram GEMMs -> LN-of-value stats -> score ->
// leaky relu -> fused softmax -> alphas.
// Grid: B * (N/16) blocks of 256 threads (8 waves). Wave w owns j-tile w.
// A-layout (16x4 f32): lanes 0-15 rows M, K={0,1}; lanes 16-31 K={2,3}.
// B-layout (4x16 f32): mirrored (lane%16 = column, lane/16 selects K-half).
// C/D (16x16 f32, 8 VGPRs): element r -> row r + 8*(lane>=16), col lane%16.
// ---------------------------------------------------------------------------
__global__ __launch_bounds__(256) void score_softmax_kernel(
    const float* __restrict__ e, const float* __restrict__ gamma,
    const float* __restrict__ beta, const float* __restrict__ att_w,
    const float* __restrict__ att_b, const float* __restrict__ qk,
    float* __restrict__ alphas) {
  __shared__ float sc[16 * N_];

  int b    = blockIdx.x >> 3;
  int i0   = (blockIdx.x & 7) << 4;
  int wave = threadIdx.x >> 5;
  int lane = threadIdx.x & 31;
  int j0   = wave << 4;
  int m    = lane & 15;           // row (A) / column (B) within tile
  int kh   = (lane >> 4) << 1;    // K half: 0 or 2

  const float* E  = e + (size_t)b * N_ * D_;
  const float* wv = att_w + 2 * D_;

  // Scalars: swg = sum(wv*gamma), wbt = sum(wv*beta)  (redundant per wave)
  float2 wv2 = ((const float2*)wv)[lane];
  float2 g2  = ((const float2*)gamma)[lane];
  float2 bt2 = ((const float2*)beta)[lane];
  float swg = wv2.x * g2.x + wv2.y * g2.y;
  float wbt = wv2.x * bt2.x + wv2.y * bt2.y;
  #pragma unroll
  for (int o = 16; o; o >>= 1) {
    swg += __shfl_xor(swg, o, 32);
    wbt += __shfl_xor(wbt, o, 32);
  }

  v8f cG = {}, cG2 = {}, cW = {};
  const float* arow = E + (i0 + m) * D_ + kh;
  const float* brow = E + (j0 + m) * D_ + kh;
  #pragma unroll
  for (int kk = 0; kk < 16; ++kk) {
    int kb = kk * 4;
    float2 ea = *(const float2*)(arow + kb);
    float2 eb = *(const float2*)(brow + kb);
    float wg0 = wv[kb + kh] * gamma[kb + kh];
    float wg1 = wv[kb + kh + 1] * gamma[kb + kh + 1];
    v2f a  = {ea.x, ea.y};
    v2f bb = {eb.x, eb.y};
    v2f a2 = {ea.x * ea.x, ea.y * ea.y};
    v2f b2 = {eb.x * eb.x, eb.y * eb.y};
    v2f aw = {ea.x * wg0, ea.y * wg1};
    cG  = __builtin_amdgcn_wmma_f32_16x16x4_f32(false, a,  false, bb,
                                                (short)0, cG,  false, false);
    cG2 = __builtin_amdgcn_wmma_f32_16x16x4_f32(false, a2, false, b2,
                                                (short)0, cG2, false, false);
    cW  = __builtin_amdgcn_wmma_f32_16x16x4_f32(false, aw, false, bb,
                                                (short)0, cW,  false, false);
  }

  // Combine: LN stats of value rows from the three Grams + q/k + bias + leaky
  float kn   = qk[B_ * N_ + b * N_ + j0 + m];
  float bias = att_b[0];
  int roff = (lane >> 4) << 3;
  #pragma unroll
  for (int r = 0; r < 8; ++r) {
    int row  = r + roff;
    float qv = qk[b * N_ + i0 + row];
    float mu   = cG[r] * (1.0f / D_);
    float var  = cG2[r] * (1.0f / D_) - mu * mu;
    float rstd = rsqrtf(var + LN_EPS_);
    float vs   = (cW[r] - mu * swg) * rstd + wbt;
    float s    = qv + kn + vs + bias;
    s = (s >= 0.0f) ? s : SLOPE_ * s;
    sc[row * N_ + j0 + m] = s;
  }
  __syncthreads();

  // Fused softmax over j: 16 threads per row (contiguous half-wave), 8 cols each
  int trow = threadIdx.x >> 4;
  int tcol = threadIdx.x & 15;
  float vals[8];
  float mx = -3.4e38f;
  #pragma unroll
  for (int c = 0; c < 8; ++c) {
    vals[c] = sc[trow * N_ + tcol + c * 16];
    mx = fmaxf(mx, vals[c]);
  }
  #pragma unroll
  for (int o = 8; o; o >>= 1) mx = fmaxf(mx, __shfl_xor(mx, o, 16));
  float sum = 0.0f;
  #pragma unroll
  for (int c = 0; c < 8; ++c) {
    vals[c] = __expf(vals[c] - mx);
    sum += vals[c];
  }
  #pragma unroll
  for (int o = 8; o; o >>= 1) sum += __shfl_xor(sum, o, 16);
  float inv = 1.0f / sum;
  float* out_row = alphas + ((size_t)(b * N_ + i0 + trow)) * N_;
  #pragma unroll
  for (int c = 0; c < 8; ++c) out_row[tcol + c * 16] = vals[c] * inv;
}

// ---------------------------------------------------------------------------
extern "C" void kernel_launch(void* const* d_in, const int* in_sizes, int n_in,
                              void* d_out, int out_size, void* d_ws,
                              size_t ws_size, hipStream_t stream) {
  const float* e     = (const float*)d_in[0];
  const float* gamma = (const float*)d_in[1];
  const float* beta  = (const float*)d_in[2];
  const float* att_w = (const float*)d_in[3];
  const float* att_b = (const float*)d_in[4];

  float* alphas = (float*)d_out;                       // B*N*N floats
  float* value  = (float*)d_out + (size_t)B_ * N_ * N_; // B*N*N*D floats
  float* qk     = (float*)d_ws;                        // 2*B*N floats

  // 1) q/k scores from LN(embeddings)
  rowstats_kernel<<<B_ * N_ / 4, 128, 0, stream>>>(e, gamma, beta, att_w, qk);

  // 2) stream the value tensor (bandwidth-bound, 134 MB)
  value_outer_kernel<<<(B_ * N_ * N_ * (D_ / 4)) / 256, 256, 0, stream>>>(e, value);

  // 3) WMMA Gram GEMMs -> score -> fused softmax -> alphas
  score_softmax_kernel<<<B_ * (N_ / 16), 256, 0, stream>>>(
      e, gamma, beta, att_w, att_b, qk, alphas);
}